// VDPMaxPooling_89541478187434
// MI455X (gfx1250) — compile-verified
//
#include <hip/hip_runtime.h>
#include <stdint.h>

// Native clang vector types (accepted by __builtin_nontemporal_store,
// lower to single b128 vmem ops).
typedef float v4f __attribute__((ext_vector_type(4)));
typedef int   v4i __attribute__((ext_vector_type(4)));

// Problem shape (fixed by reference): B=8, C=32, H=W=32 -> BC=256, HW=1024,
// pooled N = 16*16 = 256 per (b,c).
// d_out = [mu_out: 8*32*16*16 = 65536 floats][Sigma_out: 8*32*256*256 floats]
// d_ws  = argmax flat indices, 256 ints per (b,c) -> 65536 ints (256 KB).

// ---------------------------------------------------------------------------
// Kernel 1: 2x2 max-pool with flat argmax indices (matches jnp.argmax
// first-occurrence tie-breaking: window order (0,0),(0,1),(1,0),(1,1)).
// ---------------------------------------------------------------------------
__global__ __launch_bounds__(256) void vdp_pool_kernel(
    const float* __restrict__ mu, float* __restrict__ mu_out,
    int* __restrict__ idx_ws) {
  const int bc = blockIdx.x;       // 0..255
  const int p  = threadIdx.x;      // 0..255 pooled element
  const int r  = p >> 4;           // pooled row 0..15
  const int c  = p & 15;           // pooled col 0..15
  // window top-left: (2r, 2c) in a 32x32 image
  const float* base = mu + ((size_t)bc << 10) + (r << 6) + (c << 1);
  const float2 a = *(const float2*)(base);        // (2r,   2c), (2r,   2c+1)
  const float2 b = *(const float2*)(base + 32);   // (2r+1, 2c), (2r+1, 2c+1)
  float m = a.x; int w = 0;
  if (a.y > m) { m = a.y; w = 1; }
  if (b.x > m) { m = b.x; w = 2; }
  if (b.y > m) { m = b.y; w = 3; }
  const int row = (r << 1) + (w >> 1);
  const int col = (c << 1) + (w & 1);
  mu_out[(bc << 8) + p] = m;
  idx_ws[(bc << 8) + p] = (row << 5) + col;   // flat H*W index
}

// ---------------------------------------------------------------------------
// Kernel 2: Sigma double-gather. One wave32 per output row (bc, i):
//   - preload this row's 2 int4 of column indices per lane (reused -> RT)
//   - async-copy Sigma_in row idx[i] (4 KB) into this wave's private LDS
//     slice via 8x global_load_async_to_lds_b128 th:TH_LOAD_NT (row is
//     read exactly once -> don't cache it past the WGP)
//   - s_wait_asynccnt 0, gather 8 floats/lane (2x4) from LDS, non-temporal
//     b128 store (output is write-once, keep it out of L2).
// No barriers anywhere: each wave owns its row and its LDS slice.
// ---------------------------------------------------------------------------
__global__ __launch_bounds__(256) void vdp_gather_kernel(
    const float* __restrict__ Sigma, const int* __restrict__ idx_ws,
    float* __restrict__ Sigma_out) {
  __shared__ __align__(128) float lds[8][1024];   // 4 KB per wave, 32 KB/block

  const int wv   = threadIdx.x >> 5;              // wave in block: 0..7
  const int lane = threadIdx.x & 31;
  const int g    = (blockIdx.x << 3) + wv;        // global pooled-row id
  const int bc   = g >> 8;                        // (b,c) pair 0..255
  const int i    = g & 255;                       // output row 0..255

  const int* idx_bc  = idx_ws + (bc << 8);
  const int  src_row = idx_bc[i];                 // row of Sigma to gather from

  // Preload column indices now so their latency overlaps the async DMA.
  const v4i* idx4 = (const v4i*)idx_bc;           // 64 v4i = 256 idx
  const v4i c4a = idx4[lane];                     // cols for j = 4*lane..+3
  const v4i c4b = idx4[lane + 32];                // cols for j = 128+4*lane..+3

  const float* srcp = Sigma + ((size_t)bc << 20) + ((size_t)src_row << 10);
  uint64_t gaddr = (uint64_t)(uintptr_t)srcp + (uint64_t)(lane << 4);
  // Generic shared pointer carries the LDS byte address in its low 32 bits
  // (aperture in the high dword) -> truncate to get the async-load LDS addr.
  uint32_t laddr =
      (uint32_t)(uintptr_t)(&lds[wv][0]) + (uint32_t)(lane << 4);

  // 32 lanes x 16 B x 8 issues = 4 KB: the full Sigma source row into LDS.
#pragma unroll
  for (int k = 0; k < 8; ++k) {
    asm volatile("global_load_async_to_lds_b128 %0, %1, off th:TH_LOAD_NT"
                 :
                 : "v"(laddr + (uint32_t)(k << 9)),
                   "v"(gaddr + (uint64_t)(k << 9))
                 : "memory");
  }
  asm volatile("s_wait_asynccnt 0" ::: "memory");

  const float* lrow = &lds[wv][0];
  v4f* out4 = (v4f*)(Sigma_out + ((size_t)g << 8));  // 256-float row

  v4f v;
  v.x = lrow[c4a.x];
  v.y = lrow[c4a.y];
  v.z = lrow[c4a.z];
  v.w = lrow[c4a.w];
  __builtin_nontemporal_store(v, &out4[lane]);

  v.x = lrow[c4b.x];
  v.y = lrow[c4b.y];
  v.z = lrow[c4b.z];
  v.w = lrow[c4b.w];
  __builtin_nontemporal_store(v, &out4[lane + 32]);
}

// ---------------------------------------------------------------------------
extern "C" void kernel_launch(void* const* d_in, const int* in_sizes, int n_in,
                              void* d_out, int out_size, void* d_ws,
                              size_t ws_size, hipStream_t stream) {
  (void)in_sizes; (void)n_in; (void)out_size; (void)ws_size;

  const float* mu_in    = (const float*)d_in[0];  // (8,32,32,32) f32
  const float* Sigma_in = (const float*)d_in[1];  // (8,32,1024,1024) f32

  float* mu_out    = (float*)d_out;               // 65536 floats
  float* Sigma_out = (float*)d_out + 65536;       // 16777216 floats
  int*   idx_ws    = (int*)d_ws;                  // 65536 ints (256 KB)

  // 1) pool + indices: one block per (b,c)
  vdp_pool_kernel<<<256, 256, 0, stream>>>(mu_in, mu_out, idx_ws);

  // 2) gather: 65536 output rows, 8 waves (rows) per block -> 8192 blocks
  vdp_gather_kernel<<<8192, 256, 0, stream>>>(Sigma_in, idx_ws, Sigma_out);
}